// GMMConvLayer_24068996727346
// MI455X (gfx1250) — compile-verified
//
#include <hip/hip_runtime.h>
#include <hip/hip_bf16.h>

// Problem constants (from reference)
#define N_NODES 50000
#define E_EDGES 800000
#define DIN     64
#define DOUT    64
#define GDIM    3
#define KKER    8
#define KDOUT   512          // K * DOUT
#define EPS_GMM 1e-15f
#define BN_EPS  1e-5f
#define ALPHA   0.01f

typedef __attribute__((ext_vector_type(2))) float v2f;
typedef __attribute__((ext_vector_type(8))) float v8f;

// ---------------------------------------------------------------------------
// Kernel 1: x_trans[N, 512] = x[N,64] @ W_g[64,512]   (fp32 WMMA 16x16x4)
// One wave per 16x16 output tile; 3125 row tiles * 32 col tiles = 100000 waves.
// ---------------------------------------------------------------------------
__global__ void gemm_xt_kernel(const float* __restrict__ x,
                               const float* __restrict__ Wg,
                               float* __restrict__ xt) {
    const int waveId = (blockIdx.x * blockDim.x + threadIdx.x) >> 5;
    const int lane   = threadIdx.x & 31;
    const int half   = lane >> 4;     // 0: lanes 0-15, 1: lanes 16-31
    const int lm     = lane & 15;

    const int rowTile = waveId >> 5;        // 0..3124
    const int colTile = waveId & 31;        // 0..31
    const int row0 = rowTile * 16;
    const int col0 = colTile * 16;

    v8f acc = {};
#pragma unroll
    for (int kk = 0; kk < 16; ++kk) {
        const int kbase = kk * 4;
        // A fragment: M = row0+lm, K = kbase + 2*half + {0,1}
        v2f a = *(const v2f*)(x + (size_t)(row0 + lm) * DIN + kbase + 2 * half);
        // B fragment: VGPR0 -> K = kbase+2*half, VGPR1 -> K = kbase+2*half+1, N = col0+lm
        v2f b;
        b.x = Wg[(size_t)(kbase + 2 * half + 0) * KDOUT + col0 + lm];
        b.y = Wg[(size_t)(kbase + 2 * half + 1) * KDOUT + col0 + lm];
        acc = __builtin_amdgcn_wmma_f32_16x16x4_f32(
            false, a, false, b, (short)0, acc, false, false);
    }
#pragma unroll
    for (int v = 0; v < 8; ++v) {
        const int r = row0 + v + 8 * half;   // C layout: VGPR v -> M = v + 8*half
        xt[(size_t)r * KDOUT + col0 + lm] = acc[v];
    }
}

// ---------------------------------------------------------------------------
// Kernel 2: per-edge Gaussian mixture weight + gather + scatter-add.
// One wave32 per edge; lane handles 2 of the 64 output channels.
// ---------------------------------------------------------------------------
__global__ void edge_msg_kernel(const float* __restrict__ xt,
                                const int*   __restrict__ eidx,
                                const float* __restrict__ ea,
                                const float* __restrict__ mu,
                                const float* __restrict__ sigma,
                                float* __restrict__ agg,
                                float* __restrict__ cnt) {
    const int e = (blockIdx.x * blockDim.x + threadIdx.x) >> 5;
    if (e >= E_EDGES) return;                 // wave-uniform guard
    const int lane = threadIdx.x & 31;

    const int src = eidx[e];
    const int dst = eidx[E_EDGES + e];

    const float a0 = ea[(size_t)e * GDIM + 0];
    const float a1 = ea[(size_t)e * GDIM + 1];
    const float a2 = ea[(size_t)e * GDIM + 2];

    float g[KKER];
#pragma unroll
    for (int k = 0; k < KKER; ++k) {
        const float d0 = a0 - mu[k * GDIM + 0];
        const float d1 = a1 - mu[k * GDIM + 1];
        const float d2 = a2 - mu[k * GDIM + 2];
        const float s0 = sigma[k * GDIM + 0];
        const float s1 = sigma[k * GDIM + 1];
        const float s2 = sigma[k * GDIM + 2];
        const float t = -0.5f * (d0 * d0 / (EPS_GMM + s0 * s0) +
                                 d1 * d1 / (EPS_GMM + s1 * s1) +
                                 d2 * d2 / (EPS_GMM + s2 * s2));
        g[k] = __expf(t);
    }

    const float* xrow = xt + (size_t)src * KDOUT;   // [8][64] row, mostly L2-resident
    float m0 = 0.0f, m1 = 0.0f;
#pragma unroll
    for (int k = 0; k < KKER; ++k) {
        v2f v = *(const v2f*)(xrow + k * DOUT + lane * 2);
        m0 = __builtin_fmaf(g[k], v.x, m0);
        m1 = __builtin_fmaf(g[k], v.y, m1);
    }

    float* arow = agg + (size_t)dst * DOUT + lane * 2;
    atomicAdd(arow + 0, m0);
    atomicAdd(arow + 1, m1);
    if (lane == 0) atomicAdd(&cnt[dst], 1.0f);
}

// ---------------------------------------------------------------------------
// Kernel 3: out[N,64] = agg/max(cnt,1) + x @ W_root + bias   (fp32 WMMA)
// 3125 * 4 = 12500 tiles.
// ---------------------------------------------------------------------------
__global__ void gemm_out_kernel(const float* __restrict__ x,
                                const float* __restrict__ Wr,
                                const float* __restrict__ agg,
                                const float* __restrict__ cnt,
                                const float* __restrict__ bias,
                                float* __restrict__ out) {
    const int waveId = (blockIdx.x * blockDim.x + threadIdx.x) >> 5;
    if (waveId >= 3125 * 4) return;           // wave-uniform guard
    const int lane = threadIdx.x & 31;
    const int half = lane >> 4;
    const int lm   = lane & 15;

    const int rowTile = waveId >> 2;
    const int colTile = waveId & 3;
    const int row0 = rowTile * 16;
    const int col0 = colTile * 16;

    v8f acc = {};
#pragma unroll
    for (int kk = 0; kk < 16; ++kk) {
        const int kbase = kk * 4;
        v2f a = *(const v2f*)(x + (size_t)(row0 + lm) * DIN + kbase + 2 * half);
        v2f b;
        b.x = Wr[(size_t)(kbase + 2 * half + 0) * DOUT + col0 + lm];
        b.y = Wr[(size_t)(kbase + 2 * half + 1) * DOUT + col0 + lm];
        acc = __builtin_amdgcn_wmma_f32_16x16x4_f32(
            false, a, false, b, (short)0, acc, false, false);
    }

    const float bcol = bias[col0 + lm];
#pragma unroll
    for (int v = 0; v < 8; ++v) {
        const int r = row0 + v + 8 * half;
        const float inv = 1.0f / fmaxf(cnt[r], 1.0f);
        out[(size_t)r * DOUT + col0 + lm] =
            acc[v] + agg[(size_t)r * DOUT + col0 + lm] * inv + bcol;
    }
}

// ---------------------------------------------------------------------------
// Kernel 4: per-column sum / sum-of-squares for BatchNorm (atomic reduction)
// ---------------------------------------------------------------------------
__global__ void bn_reduce_kernel(const float* __restrict__ out,
                                 float* __restrict__ bnsum) {  // [64] sum, [64] sumsq
    const int tid  = blockIdx.x * blockDim.x + threadIdx.x;
    const int col  = tid & 63;
    const int r0   = tid >> 6;
    const int step = (gridDim.x * blockDim.x) >> 6;
    float s = 0.0f, s2 = 0.0f;
    for (int r = r0; r < N_NODES; r += step) {
        const float v = out[(size_t)r * DOUT + col];
        s += v;
        s2 = __builtin_fmaf(v, v, s2);
    }
    atomicAdd(&bnsum[col], s);
    atomicAdd(&bnsum[64 + col], s2);
}

// ---------------------------------------------------------------------------
// Kernel 5: BatchNorm (biased variance) + LeakyReLU, in place on d_out.
// ---------------------------------------------------------------------------
__global__ void bn_apply_kernel(float* __restrict__ out,
                                const float* __restrict__ bnsum,
                                const float* __restrict__ gamma,
                                const float* __restrict__ beta) {
    const int i = blockIdx.x * blockDim.x + threadIdx.x;
    if (i >= N_NODES * DOUT) return;
    const int col = i & 63;
    const float invN = 1.0f / (float)N_NODES;
    const float m = bnsum[col] * invN;
    const float var = bnsum[64 + col] * invN - m * m;   // biased, matches reference
    const float xv = out[i];
    const float y = gamma[col] * (xv - m) * rsqrtf(var + BN_EPS) + beta[col];
    out[i] = (y > 0.0f) ? y : ALPHA * y;
}

// ---------------------------------------------------------------------------
// Host-side launcher
// ---------------------------------------------------------------------------
extern "C" void kernel_launch(void* const* d_in, const int* in_sizes, int n_in,
                              void* d_out, int out_size, void* d_ws, size_t ws_size,
                              hipStream_t stream) {
    const float* x     = (const float*)d_in[0];
    const int*   eidx  = (const int*)  d_in[1];
    const float* ea    = (const float*)d_in[2];
    const float* Wg    = (const float*)d_in[3];
    const float* mu    = (const float*)d_in[4];
    const float* sigma = (const float*)d_in[5];
    const float* Wr    = (const float*)d_in[6];
    const float* bias  = (const float*)d_in[7];
    const float* gamma = (const float*)d_in[8];
    const float* beta  = (const float*)d_in[9];
    float* out = (float*)d_out;

    // Workspace layout (bytes):
    //   xt  : [N, 512] fp32   = 102,400,000
    //   agg : [N, 64]  fp32   =  12,800,000
    //   cnt : [N]      fp32   =     200,000
    //   bn  : [128]    fp32   =         512
    char* ws = (char*)d_ws;
    float* xt  = (float*)(ws);
    float* agg = (float*)(ws + (size_t)N_NODES * KDOUT * 4);
    float* cnt = (float*)(ws + (size_t)N_NODES * KDOUT * 4 + (size_t)N_NODES * DOUT * 4);
    float* bn  = (float*)(ws + (size_t)N_NODES * KDOUT * 4 + (size_t)N_NODES * DOUT * 4
                             + (size_t)N_NODES * 4);

    // Zero the atomic accumulation regions (agg | cnt | bn are contiguous).
    hipMemsetAsync(agg, 0,
                   (size_t)N_NODES * DOUT * 4 + (size_t)N_NODES * 4 + 128 * 4,
                   stream);

    // 1) x_trans = x @ W_g : 100000 tiles, 8 waves/block -> 12500 blocks
    gemm_xt_kernel<<<12500, 256, 0, stream>>>(x, Wg, xt);

    // 2) edge messages: 1 wave per edge, 8 edges/block -> 100000 blocks
    edge_msg_kernel<<<E_EDGES / 8, 256, 0, stream>>>(xt, eidx, ea, mu, sigma, agg, cnt);

    // 3) out = agg/cnt + x @ W_root + bias : 12500 tiles -> 1563 blocks (8 waves each)
    gemm_out_kernel<<<1563, 256, 0, stream>>>(x, Wr, agg, cnt, bias, out);

    // 4) BN stats
    bn_reduce_kernel<<<256, 256, 0, stream>>>(out, bn);

    // 5) BN + LeakyReLU in place
    bn_apply_kernel<<<(N_NODES * DOUT + 255) / 256, 256, 0, stream>>>(out, bn, gamma, beta);
}